// LSTM_Single_Layer_6794638262966
// MI455X (gfx1250) — compile-verified
//
#include <hip/hip_runtime.h>
#include <hip/hip_bf16.h>

typedef __attribute__((ext_vector_type(2))) float v2f;
typedef __attribute__((ext_vector_type(8))) float v8f;

#define HID 1024

__device__ __forceinline__ float sigmoidf_(float x) {
    return 1.0f / (1.0f + __expf(-x));
}

// ---------------------------------------------------------------------------
// GEMV over the i/g/o row-groups of a [4096 x K] fp32 weight matrix using
// V_WMMA_F32_16X16X4_F32. One 16-row tile per 256-thread block; the 8 waves
// split K into segments and partial sums are reduced in LDS.
//   out layout: [0..1023]=i rows, [1024..2047]=g rows, [2048..3071]=o rows
// ---------------------------------------------------------------------------
__global__ __launch_bounds__(256) void gemv3_wmma(const float* __restrict__ W,
                                                  const float* __restrict__ x,
                                                  float* __restrict__ out,
                                                  int K) {
    __shared__ float red[8][16];

    const int tile  = blockIdx.x;         // 0..191
    const int group = tile >> 6;          // 0:i  1:g  2:o
    const int j0    = (tile & 63) << 4;   // row offset within group
    const int wRowBase = ((group == 0) ? 0 : (group == 1) ? 2048 : 3072) + j0;

    const int tid     = threadIdx.x;
    const int lane    = tid & 31;
    const int wave    = tid >> 5;         // 0..7  (k-segment)
    const int halfSel = lane >> 4;        // 0: K pair (k,k+1)   1: (k+2,k+3)
    const int m       = lane & 15;        // A-matrix row within tile

    const int kSeg = K >> 3;              // 256 floats per wave
    const int k0   = wave * kSeg;

    const float* wptr = W + (size_t)(wRowBase + m) * (size_t)K + k0 + halfSel * 2;
    const float* xptr = x + k0 + halfSel * 2;

    v8f c = {0.f, 0.f, 0.f, 0.f, 0.f, 0.f, 0.f, 0.f};
    for (int k = 0; k < kSeg; k += 4) {
        v2f a = *(const v2f*)(wptr + k);   // A: 16x4 fp32 tile of W
        v2f b = *(const v2f*)(xptr + k);   // B: x replicated across all 16 cols
        c = __builtin_amdgcn_wmma_f32_16x16x4_f32(
                /*neg_a=*/false, a, /*neg_b=*/false, b,
                /*c_mod=*/(short)0, c, /*reuse_a=*/false, /*reuse_b=*/false);
    }

    // All N columns are identical: lane 0 holds rows 0..7 in c[0..7],
    // lane 16 holds rows 8..15 in c[0..7].
    if (m == 0) {
#pragma unroll
        for (int i = 0; i < 8; ++i) red[wave][halfSel * 8 + i] = c[i];
    }
    __syncthreads();

    if (tid < 16) {
        float s = 0.f;
#pragma unroll
        for (int w2 = 0; w2 < 8; ++w2) s += red[w2][tid];
        out[group * HID + j0 + tid] = s;
    }
}

// ---------------------------------------------------------------------------
// Activations for the map cell + the whole (tiny, K=4) pts cell.
// Produces final_in[0..1023] = h_map, final_in[1024..2047] = h_pts.
// ---------------------------------------------------------------------------
__global__ void act_map_pts(const float* __restrict__ gm,        // 3*HID raw dots
                            const float* __restrict__ b_ih_map,
                            const float* __restrict__ b_hh_map,
                            const float* __restrict__ W_ih_pts,  // [4096 x 4]
                            const float* __restrict__ b_ih_pts,
                            const float* __restrict__ b_hh_pts,
                            const float* __restrict__ goal,
                            const float* __restrict__ cur,
                            float* __restrict__ final_in) {
    int j = blockIdx.x * blockDim.x + threadIdx.x;
    if (j >= HID) return;

    // ---- map cell (h=c=0, so f-gate irrelevant) ----
    float gi = gm[j]           + b_ih_map[j]           + b_hh_map[j];
    float gg = gm[HID + j]     + b_ih_map[2 * HID + j] + b_hh_map[2 * HID + j];
    float go = gm[2 * HID + j] + b_ih_map[3 * HID + j] + b_hh_map[3 * HID + j];
    float cmap = sigmoidf_(gi) * tanhf(gg);
    final_in[j] = sigmoidf_(go) * tanhf(cmap);

    // ---- pts cell: points = [current, goal], K = 4 ----
    float p0 = cur[0], p1 = cur[1], p2 = goal[0], p3 = goal[1];
    const float* wi = W_ih_pts + (size_t)j * 4;
    const float* wg = W_ih_pts + (size_t)(2 * HID + j) * 4;
    const float* wo = W_ih_pts + (size_t)(3 * HID + j) * 4;
    float pi = wi[0]*p0 + wi[1]*p1 + wi[2]*p2 + wi[3]*p3 + b_ih_pts[j]           + b_hh_pts[j];
    float pg = wg[0]*p0 + wg[1]*p1 + wg[2]*p2 + wg[3]*p3 + b_ih_pts[2 * HID + j] + b_hh_pts[2 * HID + j];
    float po = wo[0]*p0 + wo[1]*p1 + wo[2]*p2 + wo[3]*p3 + b_ih_pts[3 * HID + j] + b_hh_pts[3 * HID + j];
    float cp = sigmoidf_(pi) * tanhf(pg);
    final_in[HID + j] = sigmoidf_(po) * tanhf(cp);
}

// ---------------------------------------------------------------------------
// lstm1 activations + fc (H -> 2) in a single block.
// ---------------------------------------------------------------------------
__global__ __launch_bounds__(256) void final_fc(const float* __restrict__ g1,  // 3*HID raw dots
                                                const float* __restrict__ b_ih_1,
                                                const float* __restrict__ b_hh_1,
                                                const float* __restrict__ fc_w,  // [2 x HID]
                                                const float* __restrict__ fc_b,
                                                float* __restrict__ out) {
    __shared__ float h1[HID];
    __shared__ float r0[256];
    __shared__ float r1[256];
    const int tid = threadIdx.x;

    for (int j = tid; j < HID; j += 256) {
        float gi = g1[j]           + b_ih_1[j]           + b_hh_1[j];
        float gg = g1[HID + j]     + b_ih_1[2 * HID + j] + b_hh_1[2 * HID + j];
        float go = g1[2 * HID + j] + b_ih_1[3 * HID + j] + b_hh_1[3 * HID + j];
        float c  = sigmoidf_(gi) * tanhf(gg);
        h1[j] = sigmoidf_(go) * tanhf(c);
    }
    __syncthreads();

    float a0 = 0.f, a1 = 0.f;
    for (int j = tid; j < HID; j += 256) {
        float h = h1[j];
        a0 += h * fc_w[j];
        a1 += h * fc_w[HID + j];
    }
    r0[tid] = a0;
    r1[tid] = a1;
    __syncthreads();
    for (int s = 128; s > 0; s >>= 1) {
        if (tid < s) { r0[tid] += r0[tid + s]; r1[tid] += r1[tid + s]; }
        __syncthreads();
    }
    if (tid == 0) {
        out[0] = r0[0] + fc_b[0];
        out[1] = r1[0] + fc_b[1];
    }
}

// ---------------------------------------------------------------------------
// Launch: 4 kernels on `stream`; stream order provides the LSTM dependency
// sync. W_hh_* matrices and f-gate rows are intentionally never read
// (multiplied by zero state) -> total HBM traffic ~48 MB, ~2.1 us floor.
// ---------------------------------------------------------------------------
extern "C" void kernel_launch(void* const* d_in, const int* in_sizes, int n_in,
                              void* d_out, int out_size, void* d_ws, size_t ws_size,
                              hipStream_t stream) {
    const float* goal     = (const float*)d_in[0];
    const float* cur      = (const float*)d_in[1];
    const float* map      = (const float*)d_in[2];
    const float* W_ih_map = (const float*)d_in[3];
    const float* b_ih_map = (const float*)d_in[5];
    const float* b_hh_map = (const float*)d_in[6];
    const float* W_ih_pts = (const float*)d_in[7];
    const float* b_ih_pts = (const float*)d_in[9];
    const float* b_hh_pts = (const float*)d_in[10];
    const float* W_ih_1   = (const float*)d_in[11];
    const float* b_ih_1   = (const float*)d_in[13];
    const float* b_hh_1   = (const float*)d_in[14];
    const float* fc_w     = (const float*)d_in[15];
    const float* fc_b     = (const float*)d_in[16];

    float* ws  = (float*)d_ws;
    float* gm  = ws;            // 3*HID raw map-cell gate dots
    float* fin = ws + 4096;     // 2*HID concat(h_map, h_pts)
    float* g1  = ws + 8192;     // 3*HID raw lstm1 gate dots

    gemv3_wmma<<<192, 256, 0, stream>>>(W_ih_map, map, gm, 2048);
    act_map_pts<<<4, 256, 0, stream>>>(gm, b_ih_map, b_hh_map,
                                       W_ih_pts, b_ih_pts, b_hh_pts,
                                       goal, cur, fin);
    gemv3_wmma<<<192, 256, 0, stream>>>(W_ih_1, fin, g1, 2048);
    final_fc<<<1, 256, 0, stream>>>(g1, b_ih_1, b_hh_1, fc_w, fc_b, (float*)d_out);
}